// AttentionHead_13469017441038
// MI455X (gfx1250) — compile-verified
//
#include <hip/hip_runtime.h>

typedef __attribute__((ext_vector_type(2))) float v2f;
typedef __attribute__((ext_vector_type(8))) float v8f;

#define B_ 2048
#define K_ 1024
#define D_ 64
#define C_ 32
#define WAVES 4
#define TPB (WAVES * 32)
#define KEYS_PER_WAVE (K_ / WAVES)          // 256
#define TILES_PER_WAVE (KEYS_PER_WAVE / 16) // 16
#define ROWPAD 68                            // 64 + 4 floats: conflict-free ds_load_b64

// Async-DMA one 16x64 f32 key tile (4 KB) from global into a padded LDS buffer.
// 8 x global_load_async_to_lds_b128: each lane moves 16B, tracked by ASYNCcnt.
__device__ __forceinline__ void stage_tile_async(const float* __restrict__ src,
                                                 unsigned ldsbase, int lane) {
    #pragma unroll
    for (int q = 0; q < 8; ++q) {
        const int idx = q * 128 + lane * 4;      // flat float index in 16x64 tile
        const int row = idx >> 6, cc = idx & 63;
        unsigned loff = ldsbase + (unsigned)((row * ROWPAD + cc) * 4);
        unsigned long long ga = (unsigned long long)(const void*)(src + idx);
        asm volatile("global_load_async_to_lds_b128 %0, %1, off"
                     :: "v"(loff), "v"(ga) : "memory");
    }
}

__global__ __launch_bounds__(TPB) void nw_attention_kernel(
    const float* __restrict__ x, const float* __restrict__ keys,
    const float* __restrict__ values, const float* __restrict__ gamma,
    float* __restrict__ out)
{
    __shared__ __align__(16) float lds_x[D_];
    __shared__ __align__(16) float lds_keys[WAVES][2][16 * ROWPAD]; // double buffer
    __shared__ float lds_scores[K_];
    __shared__ float red_max[WAVES];
    __shared__ float red_sum[WAVES];
    __shared__ float red_acc[WAVES][C_];

    const int b    = blockIdx.x;
    const int tid  = threadIdx.x;
    const int w    = tid >> 5;
    const int lane = tid & 31;
    const int half = lane >> 4;   // 0: K-slots {0,1}, 1: K-slots {2,3}
    const int col  = lane & 15;   // B-matrix column / key index in tile

    const float g = gamma[0];

    // ---- stage x[b] into LDS ----
    if (tid < D_) lds_x[tid] = x[(size_t)b * D_ + tid];
    __syncthreads();

    // ||x||^2 (same in every lane; broadcast LDS reads)
    float xnorm = 0.f;
    #pragma unroll
    for (int d = 0; d < D_; d += 4) {
        float4 t = *reinterpret_cast<const float4*>(&lds_x[d]);
        xnorm += t.x * t.x + t.y * t.y + t.z * t.z + t.w * t.w;
    }

    // ---- preload A operands: x broadcast across all 16 rows, per K-chunk ----
    // A(16x4) f32 layout: lanes 0-15 hold K=0,1 in v[0],v[1]; lanes 16-31 hold K=2,3.
    v2f a[16];
    #pragma unroll
    for (int c = 0; c < 16; ++c) {
        int dbase = 4 * c + 2 * half;
        a[c] = *reinterpret_cast<const v2f*>(&lds_x[dbase]);
    }

    // =====================  Pass 1: scores  =====================
    float lmax = -3.0e38f;
    const float* keyb = keys + (size_t)b * K_ * D_;
    const unsigned lb0 = (unsigned)(unsigned long long)(const void*)lds_keys[w][0];
    const unsigned lb1 = (unsigned)(unsigned long long)(const void*)lds_keys[w][1];

    // prime the pipeline: tile 0 -> buffer 0
    stage_tile_async(keyb + (size_t)(w * KEYS_PER_WAVE) * D_, lb0, lane);

    for (int t = 0; t < TILES_PER_WAVE; ++t) {
        const int k0 = w * KEYS_PER_WAVE + t * 16;

        // overlap: kick off next tile's DMA into the other buffer, then wait
        // only for the current tile's 8 async ops (in-order completion).
        if (t + 1 < TILES_PER_WAVE) {
            stage_tile_async(keyb + (size_t)(k0 + 16) * D_,
                             ((t + 1) & 1) ? lb1 : lb0, lane);
            asm volatile("s_wait_asynccnt 0x8" ::: "memory");
        } else {
            asm volatile("s_wait_asynccnt 0x0" ::: "memory");
        }

        const float* cbuf = lds_keys[w][t & 1];

        // B(4x16): column j lives in lanes j and j+16; ds_load_b64 per chunk
        v8f acc = {};
        float pnorm = 0.f;
        #pragma unroll
        for (int c = 0; c < 16; ++c) {
            int off = col * ROWPAD + 4 * c + 2 * half;
            v2f kb = *reinterpret_cast<const v2f*>(&cbuf[off]);
            pnorm += kb.x * kb.x + kb.y * kb.y;
            acc = __builtin_amdgcn_wmma_f32_16x16x4_f32(
                false, a[c], false, kb, (short)0, acc, false, false);
        }
        // dot(key col) replicated across all rows -> element 0 of C/D
        float dot   = acc[0];
        float knorm = pnorm + __shfl_xor(pnorm, 16, 32);
        float s     = -g * (xnorm + knorm - 2.f * dot);

        if (half == 0) lds_scores[k0 + col] = s;
        lmax = fmaxf(lmax, s);
    }

    // wave max -> block max
    #pragma unroll
    for (int m = 16; m >= 1; m >>= 1) lmax = fmaxf(lmax, __shfl_xor(lmax, m, 32));
    if (lane == 0) red_max[w] = lmax;
    __syncthreads();

    float gmax = red_max[0];
    #pragma unroll
    for (int i = 1; i < WAVES; ++i) gmax = fmaxf(gmax, red_max[i]);

    // ==============  Pass 2: softmax-weighted value sum  ==============
    const float* valb = values + (size_t)b * K_ * C_;
    float accv = 0.f;   // lane = output channel
    float sume = 0.f;
    const int kbeg = w * KEYS_PER_WAVE;
    #pragma unroll 4
    for (int k = kbeg; k < kbeg + KEYS_PER_WAVE; ++k) {
        float wk = __expf(lds_scores[k] - gmax);   // broadcast LDS read
        float vv = valb[(size_t)k * C_ + lane];    // 128B coalesced per wave
        accv += wk * vv;
        sume += wk;
    }
    red_acc[w][lane] = accv;
    if (lane == 0) red_sum[w] = sume;
    __syncthreads();

    if (w == 0) {
        float tot = 0.f, den = 0.f;
        #pragma unroll
        for (int i = 0; i < WAVES; ++i) { tot += red_acc[i][lane]; den += red_sum[i]; }
        out[(size_t)b * C_ + lane] = tot / den;
    }
}

extern "C" void kernel_launch(void* const* d_in, const int* in_sizes, int n_in,
                              void* d_out, int out_size, void* d_ws, size_t ws_size,
                              hipStream_t stream) {
    const float* x      = (const float*)d_in[0];
    const float* keys   = (const float*)d_in[1];
    const float* values = (const float*)d_in[2];
    const float* gamma  = (const float*)d_in[3];
    float* out          = (float*)d_out;
    nw_attention_kernel<<<dim3(B_), dim3(TPB), 0, stream>>>(x, keys, values, gamma, out);
}